// ResampleFeatureMap_64295660421504
// MI455X (gfx1250) — compile-verified
//
#include <hip/hip_runtime.h>

typedef __attribute__((ext_vector_type(2))) float v2f;
typedef __attribute__((ext_vector_type(8))) float v8f;

#define WAVES  8
#define BLOCK  (WAVES * 32)
#define CHUNK  1024          // keys staged per LDS chunk (16 KB as float4), x2 buffers

// branchless sorted-triple insert: keeps 3 smallest u64 keys (k0<=k1<=k2)
__device__ __forceinline__ void key3_insert(unsigned long long key,
                                            unsigned long long& k0,
                                            unsigned long long& k1,
                                            unsigned long long& k2) {
  unsigned long long lo0 = (k0 < key) ? k0 : key;
  unsigned long long hi0 = (k0 < key) ? key : k0;
  k0 = lo0;
  unsigned long long lo1 = (k1 < hi0) ? k1 : hi0;
  unsigned long long hi1 = (k1 < hi0) ? hi0 : k1;
  k1 = lo1;
  k2 = (k2 < hi1) ? k2 : hi1;
}

__device__ __forceinline__ unsigned long long shfl_xor_u64(unsigned long long v, int m) {
  int lo = (int)(unsigned int)v;
  int hi = (int)(unsigned int)(v >> 32);
  lo = __shfl_xor(lo, m, 32);
  hi = __shfl_xor(hi, m, 32);
  return ((unsigned long long)(unsigned int)hi << 32) | (unsigned int)lo;
}

__global__ __launch_bounds__(BLOCK)
void three_interp_wmma_kernel(const float* __restrict__ xyz,
                              const float* __restrict__ new_xyz,
                              const float* __restrict__ features,
                              float* __restrict__ out,
                              int Mb, int Nb, int Bc, int C) {
  // keys staged as (x, y, z, |k|^2) so the B-fragment is one ds_load_b64
  __shared__ __align__(16) float4 skey[2][CHUNK];

  const int tid  = threadIdx.x;
  const int lane = tid & 31;
  const int wave = tid >> 5;
  const int col  = lane & 15;        // N-column class this lane owns in WMMA D
  const int half = lane >> 4;        // 0: rows 0-7 / K=0,1 ; 1: rows 8-15 / K=2,3

  const int Ntot  = Bc * Nb;
  const int qtile = blockIdx.x * WAVES + wave;   // 16 queries per wave
  const int qbase = qtile * 16;
  int batch = qbase / Nb; if (batch > Bc - 1) batch = Bc - 1;
  const int kbase = batch * Mb;                  // global key-row base for batch

  // ---------------- A operand: 16 queries x K=4 = (-2x, -2y, -2z, 1) -------
  // f32 A 16x4 layout: VGPR0 lanes0-15 K=0, lanes16-31 K=2; VGPR1 K=1 / K=3.
  int qidx = qbase + col; if (qidx >= Ntot) qidx = Ntot - 1;
  const float qx = new_xyz[(size_t)qidx * 3 + 0];
  const float qy = new_xyz[(size_t)qidx * 3 + 1];
  const float qz = new_xyz[(size_t)qidx * 3 + 2];
  const float qn2 = qx * qx + qy * qy + qz * qz;
  v2f a;
  a.x = half ? (-2.0f * qz) : (-2.0f * qx);
  a.y = half ? 1.0f         : (-2.0f * qy);

  // C operand: |q|^2 of the 8 rows this lane holds -> WMMA emits true d^2 >= 0
  v8f cvec;
#pragma unroll
  for (int r = 0; r < 8; ++r) cvec[r] = __shfl(qn2, half * 8 + r, 32);

  // per-lane running top-3 (packed sortable keys) for the 8 rows this lane holds
  unsigned long long k0[8], k1[8], k2[8];
  const unsigned long long KINIT =
      0xFFFFFFFF00000000ull | (unsigned int)kbase;   // max score, valid id
#pragma unroll
  for (int r = 0; r < 8; ++r) { k0[r] = k1[r] = k2[r] = KINIT; }

  const float* kptr = xyz + (size_t)kbase * 3;
  const int nch = (Mb + CHUNK - 1) / CHUNK;

  // ---- staging helpers: thread t owns keys [4t, 4t+4) of each chunk ----
  auto load_quad = [&](int ch, float4* p /*[3]*/) {
    const int gf = (ch * CHUNK + tid * 4) * 3;       // first float of 4 keys
    if (gf + 11 < Mb * 3) {
      const float4* s4 = (const float4*)(kptr + gf);
      p[0] = s4[0]; p[1] = s4[1]; p[2] = s4[2];
    } else {
      float tmp[12];
#pragma unroll
      for (int i = 0; i < 12; ++i)
        tmp[i] = (gf + i < Mb * 3) ? kptr[gf + i] : 1.0e18f;
      p[0] = make_float4(tmp[0], tmp[1], tmp[2], tmp[3]);
      p[1] = make_float4(tmp[4], tmp[5], tmp[6], tmp[7]);
      p[2] = make_float4(tmp[8], tmp[9], tmp[10], tmp[11]);
    }
  };
  auto store_quad = [&](int buf, const float4* p /*[3]*/) {
    const float kxv[4] = {p[0].x, p[0].w, p[1].z, p[2].y};
    const float kyv[4] = {p[0].y, p[1].x, p[1].w, p[2].z};
    const float kzv[4] = {p[0].z, p[1].y, p[2].x, p[2].w};
#pragma unroll
    for (int j = 0; j < 4; ++j) {
      const float s2 = kxv[j] * kxv[j] + kyv[j] * kyv[j] + kzv[j] * kzv[j];
      skey[buf][tid * 4 + j] = make_float4(kxv[j], kyv[j], kzv[j], s2);
    }
  };

  {
    float4 p[3];
    load_quad(0, p);
    store_quad(0, p);
  }
  __syncthreads();

  for (int ch = 0; ch < nch; ++ch) {
    const int cur   = ch & 1;
    const bool more = (ch + 1) < nch;

    // global->reg prefetch of next chunk (overlaps with compute below)
    float4 pre[3];
    if (more) {
      load_quad(ch + 1, pre);
      if (ch + 2 < nch)
        __builtin_prefetch(kptr + (size_t)(ch + 2) * CHUNK * 3 + tid * 4, 0, 1);
    }

    // ---- score 16 queries x CHUNK keys via WMMA f32 16x16x4 ----
    // lane-private B fragment address: lanes0-15 -> (x,y), lanes16-31 -> (z,|k|^2)
    const float2* bptr =
        (const float2*)((const char*)&skey[cur][0] + col * 16 + half * 8);
    const unsigned int idbase = (unsigned int)(kbase + ch * CHUNK + col);
#pragma unroll 2
    for (int t = 0; t < CHUNK / 16; ++t) {
      const float2 kv = bptr[t * 32];        // ds_load_b64, imm offset t*256
      v2f b; b.x = kv.x; b.y = kv.y;
      // d^2(M,N) = |q_M|^2 - 2 q_M . k_N + |k_N|^2  (C operand carries |q|^2)
      v8f acc = __builtin_amdgcn_wmma_f32_16x16x4_f32(
          /*neg_a=*/false, a, /*neg_b=*/false, b,
          /*c_mod=*/(short)0, cvec, /*reuse_a=*/false, /*reuse_b=*/false);
      const unsigned int id = idbase + (unsigned int)(t * 16);
#pragma unroll
      for (int r = 0; r < 8; ++r) {
        // single v_med3_f32: clamp tiny cancellation negatives to 0
        const float s = __builtin_amdgcn_fmed3f(acc[r], 0.0f, 3.402823466e+38f);
        unsigned long long key =
            ((unsigned long long)__float_as_uint(s) << 32) | id;
        key3_insert(key, k0[r], k1[r], k2[r]);
      }
    }

    __syncthreads();
    if (more) store_quad(cur ^ 1, pre);
    __syncthreads();
  }

  // ---- merge per-column-class top-3 across the 16 lanes of each half ----
#pragma unroll
  for (int m = 1; m <= 8; m <<= 1) {
#pragma unroll
    for (int r = 0; r < 8; ++r) {
      unsigned long long o0 = shfl_xor_u64(k0[r], m);
      unsigned long long o1 = shfl_xor_u64(k1[r], m);
      unsigned long long o2 = shfl_xor_u64(k2[r], m);
      key3_insert(o0, k0[r], k1[r], k2[r]);
      key3_insert(o1, k0[r], k1[r], k2[r]);
      key3_insert(o2, k0[r], k1[r], k2[r]);
    }
  }

  // ---- weights + fused feature interpolation ----
  // lanes 0-15 all hold final rows 0-7; lanes 16-31 hold rows 8-15.
#pragma unroll
  for (int r = 0; r < 8; ++r) {
    const int q = qbase + half * 8 + r;
    const float d0 = sqrtf(__uint_as_float((unsigned int)(k0[r] >> 32)));
    const float d1 = sqrtf(__uint_as_float((unsigned int)(k1[r] >> 32)));
    const float d2 = sqrtf(__uint_as_float((unsigned int)(k2[r] >> 32)));
    const int i0 = (int)(unsigned int)k0[r];
    const int i1 = (int)(unsigned int)k1[r];
    const int i2 = (int)(unsigned int)k2[r];
    float w0 = 1.0f / (d0 + 1e-8f);
    float w1 = 1.0f / (d1 + 1e-8f);
    float w2 = 1.0f / (d2 + 1e-8f);
    const float inv = 1.0f / (w0 + w1 + w2);
    w0 *= inv; w1 *= inv; w2 *= inv;

    if (q < Ntot) {
      const float* f0 = features + (size_t)i0 * C;
      const float* f1 = features + (size_t)i1 * C;
      const float* f2 = features + (size_t)i2 * C;
      float* op = out + (size_t)q * C;
      // 16 lanes per half cover channels in float2 strides (coalesced)
      for (int c = col * 2; c + 1 < C; c += 32) {
        const float2 v0 = *(const float2*)(f0 + c);
        const float2 v1 = *(const float2*)(f1 + c);
        const float2 v2 = *(const float2*)(f2 + c);
        float2 o;
        o.x = w0 * v0.x + w1 * v1.x + w2 * v2.x;
        o.y = w0 * v0.y + w1 * v1.y + w2 * v2.y;
        *(float2*)(op + c) = o;
      }
    }
  }
}

extern "C" void kernel_launch(void* const* d_in, const int* in_sizes, int n_in,
                              void* d_out, int out_size, void* d_ws, size_t ws_size,
                              hipStream_t stream) {
  const float* xyz      = (const float*)d_in[0];
  const float* new_xyz  = (const float*)d_in[2];
  const float* features = (const float*)d_in[4];
  float* out = (float*)d_out;

  const int B  = in_sizes[1];                 // xyz_batch_cnt length
  const int Mb = in_sizes[0] / (3 * B);
  const int Nb = in_sizes[2] / (3 * B);
  const int C  = in_sizes[4] / (B * Mb);

  const int Ntot   = B * Nb;
  const int qtiles = (Ntot + 15) / 16;
  const int blocks = (qtiles + WAVES - 1) / WAVES;

  three_interp_wmma_kernel<<<blocks, BLOCK, 0, stream>>>(
      xyz, new_xyz, features, out, Mb, Nb, B, C);
}